// PartChamferLoss_Brute_3204045603272
// MI455X (gfx1250) — compile-verified
//
#include <hip/hip_runtime.h>
#include <cstdint>
#include <cstddef>

typedef __attribute__((ext_vector_type(2))) float v2f;
typedef __attribute__((ext_vector_type(8))) float v8f;

namespace {

constexpr int Bb = 16, Kk = 8, Mm = 1024, Nn = 1024;
constexpr int BKS = Bb * Kk;     // 128 (b,k) pairs
constexpr int STRIPS = 8;        // 8 strips of 128 rows per (b,k)
constexpr int TPB = 256;         // 8 wave32s

__global__ __launch_bounds__(256) void init_colmin(unsigned* colmin, int n) {
  int i = blockIdx.x * blockDim.x + threadIdx.x;
  if (i < n) colmin[i] = 0x7F800000u;  // +inf bits
}

// One block = one (b,k) pair x one 128-row strip. Each wave owns a 16-row
// A-tile and sweeps all 64 n-tiles with V_WMMA_F32_16X16X4_F32:
//   A row  = [x_m, y_m, z_m, 1]
//   B col  = [-2x_n, -2y_n, -2z_n, |d_n|^2]
//   C      = |s_m|^2 per row
//   D      = squared distance tile (sqrt deferred to the final reduction).
__global__ __launch_bounds__(256) void chamfer_tiles(const float* __restrict__ src,
                                                     const float* __restrict__ dst,
                                                     float* __restrict__ rowmin,
                                                     unsigned* __restrict__ colmin) {
  // +1 padding tile so the software-pipelined prefetch can read one tile past
  // the end unconditionally (value never consumed).
  __shared__ float ldsB[(Nn + 16) * 4];  // per n: {-2x,-2y,-2z,|d|^2} (16.25 KB)
  __shared__ unsigned ldsCol[Nn];        // per-block col-min (f32 bits) (4 KB)

  const int bk    = blockIdx.x >> 3;
  const int strip = blockIdx.x & 7;
  const float* __restrict__ S = src + (size_t)bk * 3 * Mm;
  const float* __restrict__ D = dst + (size_t)bk * 3 * Nn;
  const int t = threadIdx.x;

  // Stage dst-side tile data into LDS once per block.
  for (int n = t; n < Nn; n += TPB) {
    float x = D[n], y = D[Nn + n], z = D[2 * Nn + n];
    float4 v;
    v.x = -2.0f * x; v.y = -2.0f * y; v.z = -2.0f * z;
    v.w = x * x + y * y + z * z;
    *(float4*)&ldsB[n * 4] = v;          // ds_store_b128
    ldsCol[n] = 0x7F800000u;
  }
  __syncthreads();

  const int lane   = t & 31;
  const int wave   = t >> 5;
  const int lane16 = lane & 15;
  const int hi     = lane >> 4;          // lane half selects K={0,1} vs K={2,3}
  const int mbase  = strip * 128 + wave * 16;
  const int m      = mbase + lane16;

  // A-matrix 16x4 f32 (2 VGPRs): lanes 0-15 hold (x,y), lanes 16-31 hold (z,1).
  float a0   = S[(hi ? 2 * Mm : 0) + m];   // x or z
  float ytmp = S[Mm + m];                  // y (loaded by all lanes, coalesced)
  float a1   = hi ? 1.0f : ytmp;
  v2f a = {a0, a1};

  // ||s_m||^2 per row -> C matrix (row-constant), built once via shuffles.
  float zpart = __shfl_xor(a0, 16, 32);    // lo lanes receive z of their m
  float sq = a0 * a0 + ytmp * ytmp + zpart * zpart;  // valid in lanes 0-15
  v8f c;
#pragma unroll
  for (int r = 0; r < 8; ++r)
    c[r] = __shfl(sq, r + hi * 8, 32);     // C/D row M = r (+8 in hi half)

  v8f rmin;
#pragma unroll
  for (int r = 0; r < 8; ++r) rmin[r] = __builtin_inff();

  const int ldsIdx = lane16 * 4 + hi * 2;  // float2 per lane: (K0,K1) or (K2,K3)

  // Hot loop: 64 n-tiles, software-pipelined one tile ahead so the ds_load for
  // tile nt+1 is in flight while tile nt runs WMMA + min reduction.
  float2 bb = *(const float2*)&ldsB[ldsIdx];
#pragma unroll 4
  for (int nt = 0; nt < Nn / 16; ++nt) {
    const float2 bbn = *(const float2*)&ldsB[(nt + 1) * 64 + ldsIdx];  // prefetch

    v2f b = {bb.x, bb.y};
    v8f d = __builtin_amdgcn_wmma_f32_16x16x4_f32(
        false, a, false, b, (short)0, c, false, false);

#pragma unroll
    for (int r = 0; r < 8; ++r)
      rmin[r] = fminf(rmin[r], d[r]);      // fuses to v_min3 across unroll

    // Per-lane min over the 8 rows: med3(-inf) avoids IEEE canonicalize ops,
    // final fminf+fmaxf fuses into v_minmax_num (clamp makes uint order valid).
    float cm = __builtin_amdgcn_fmed3f(d[0], d[1], -__builtin_inff());
    cm = fminf(fminf(cm, d[2]), d[3]);
    cm = fminf(fminf(cm, d[4]), d[5]);
    cm = fminf(cm, d[6]);
    cm = fmaxf(fminf(cm, d[7]), 0.0f);

    // Both lane halves hold the same n: all 32 lanes min into the shared slot.
    atomicMin(&ldsCol[nt * 16 + lane16], __float_as_uint(cm));

    bb = bbn;
  }

  // Row-min: reduce across the 16 column lanes of each half.
#pragma unroll
  for (int mask = 1; mask < 16; mask <<= 1) {
#pragma unroll
    for (int r = 0; r < 8; ++r)
      rmin[r] = fminf(rmin[r], __shfl_xor(rmin[r], mask, 32));
  }
  if (lane16 == 0) {
#pragma unroll
    for (int r = 0; r < 8; ++r)
      rowmin[(size_t)bk * Mm + mbase + hi * 8 + r] = fmaxf(rmin[r], 0.0f);
  }

  __syncthreads();
  for (int n = t; n < Nn; n += TPB)
    atomicMin(&colmin[(size_t)bk * Nn + n], ldsCol[n]);
}

// Deterministic two-stage reduction: sqrt the 2*131072 min values and sum.
__global__ __launch_bounds__(256) void reduce_stage1(const float* __restrict__ rowmin,
                                                     const unsigned* __restrict__ colmin,
                                                     float* __restrict__ partial) {
  __shared__ float sm[256];
  const int t = threadIdx.x;
  const int base = blockIdx.x * 512;     // 256 blocks x 512 covers 131072
  float s = 0.0f;
  for (int i = t; i < 512; i += 256) {
    s += sqrtf(rowmin[base + i]);
    s += sqrtf(__uint_as_float(colmin[base + i]));
  }
  sm[t] = s;
  __syncthreads();
  for (int off = 128; off > 0; off >>= 1) {
    if (t < off) sm[t] += sm[t + off];
    __syncthreads();
  }
  if (t == 0) partial[blockIdx.x] = sm[0];
}

__global__ __launch_bounds__(256) void reduce_stage2(const float* __restrict__ partial,
                                                     float* __restrict__ out) {
  __shared__ float sm[256];
  const int t = threadIdx.x;
  sm[t] = partial[t];
  __syncthreads();
  for (int off = 128; off > 0; off >>= 1) {
    if (t < off) sm[t] += sm[t + off];
    __syncthreads();
  }
  // mean_k(fwd+bwd) with uniform weights = total / (K*B*M)  (M==N)
  if (t == 0) out[0] = sm[0] * (1.0f / (float)(Bb * Kk * Mm));
}

}  // namespace

extern "C" void kernel_launch(void* const* d_in, const int* in_sizes, int n_in,
                              void* d_out, int out_size, void* d_ws, size_t ws_size,
                              hipStream_t stream) {
  (void)in_sizes; (void)n_in; (void)out_size; (void)ws_size;
  const float* src = (const float*)d_in[0];
  const float* dst = (const float*)d_in[1];
  char* ws = (char*)d_ws;
  float*    rowmin  = (float*)ws;                                          // 512 KB
  unsigned* colmin  = (unsigned*)(ws + (size_t)BKS * Mm * sizeof(float));  // 512 KB
  float*    partial = (float*)(ws + 2 * (size_t)BKS * Mm * sizeof(float)); // 1 KB
  float* out = (float*)d_out;

  init_colmin<<<(BKS * Nn + 255) / 256, 256, 0, stream>>>(colmin, BKS * Nn);
  chamfer_tiles<<<BKS * STRIPS, TPB, 0, stream>>>(src, dst, rowmin, colmin);
  reduce_stage1<<<256, 256, 0, stream>>>(rowmin, colmin, partial);
  reduce_stage2<<<1, 256, 0, stream>>>(partial, out);
}